// LengthRegulator_71210557768149
// MI455X (gfx1250) — compile-verified
//
#include <hip/hip_runtime.h>
#include <stdint.h>

// Length regulator: B=32 rows, L=1024 tokens, D=512 feat, T=4096 out frames.
// Stage 1: inclusive scan of durations per row -> cum[B][L] (int32) in d_ws.
// Stage 2: per output frame, binary-search cum, copy D floats (or zeros).

typedef float v4f __attribute__((ext_vector_type(4)));
typedef int   v4i __attribute__((vector_size(4 * sizeof(int))));

#define AS3 __attribute__((address_space(3)))
#define AS1 __attribute__((address_space(1)))

#if __has_builtin(__builtin_amdgcn_global_load_async_to_lds_b128) && \
    __has_builtin(__builtin_amdgcn_global_store_async_from_lds_b128) && \
    __has_builtin(__builtin_amdgcn_s_wait_asynccnt)
#define LR_HAVE_ASYNC 1
#else
#define LR_HAVE_ASYNC 0
#endif

// ---------------- Stage 2: gather / expand (listed first for disasm) -------
// Grid = (T, B); one block (D/4 = 128 threads) per output frame (b, t).
// Thread 0 does searchsorted(cum_row, t, 'right'); every thread then moves
// one 16B chunk via the ASYNCcnt global->LDS->global DMA path.
__global__ void lr_gather_kernel(const float* __restrict__ x,
                                 const int* __restrict__ cum,
                                 float* __restrict__ out,
                                 int L, int D, int T) {
  __shared__ int s_idx;
#if LR_HAVE_ASYNC
  __shared__ float buf[512];
#endif
  const int t = blockIdx.x;          // output frame index
  const int b = blockIdx.y;          // batch row
  const int* crow = cum + (size_t)b * L;

  if (threadIdx.x == 0) {
    int idx = -1;
    if (t < crow[L - 1]) {           // frame inside total duration?
      int lo = 0, hi = L;            // first i with crow[i] > t
      while (lo < hi) {
        int mid = (lo + hi) >> 1;
        if (crow[mid] <= t) lo = mid + 1; else hi = mid;
      }
      idx = lo;                      // < L guaranteed since t < crow[L-1]
    }
    s_idx = idx;
  }
  __syncthreads();

  const int idx  = s_idx;
  const int lane = threadIdx.x;      // 0 .. D/4-1
  float* dst = out + ((size_t)b * T + t) * (size_t)D + lane * 4;

#if LR_HAVE_ASYNC
  if (idx >= 0) {
    const float* src = x + ((size_t)b * L + idx) * (size_t)D + lane * 4;
    // global -> LDS (ASYNCcnt DMA path)
    __builtin_amdgcn_global_load_async_to_lds_b128(
        (AS1 v4i*)(src), (AS3 v4i*)(&buf[lane * 4]), 0, 0);
    __builtin_amdgcn_s_wait_asynccnt(0);
    // LDS -> global; drains via implicit wait-idle at s_endpgm
    __builtin_amdgcn_global_store_async_from_lds_b128(
        (AS1 v4i*)(dst), (AS3 v4i*)(&buf[lane * 4]), 0, 0);
  } else {
    v4f z = {0.f, 0.f, 0.f, 0.f};
    __builtin_nontemporal_store(z, (v4f*)dst);
  }
#else
  v4f v = {0.f, 0.f, 0.f, 0.f};
  if (idx >= 0) {
    const v4f* src = (const v4f*)(x + ((size_t)b * L + idx) * (size_t)D);
    v = src[lane];
  }
  __builtin_nontemporal_store(v, (v4f*)dst);   // write-once stream
#endif
}

// ---------------- Stage 1: per-row inclusive scan (Hillis-Steele in LDS) ----
__global__ void lr_scan_kernel(const int* __restrict__ dur,
                               int* __restrict__ cum, int L) {
  __shared__ int s[1024];
  const int b = blockIdx.x;
  const int t = threadIdx.x;
  int d = dur[(size_t)b * L + t];
  s[t] = d > 0 ? d : 0;
  __syncthreads();
  for (int off = 1; off < L; off <<= 1) {
    int add = (t >= off) ? s[t - off] : 0;
    __syncthreads();
    s[t] += add;
    __syncthreads();
  }
  cum[(size_t)b * L + t] = s[t];
}

extern "C" void kernel_launch(void* const* d_in, const int* in_sizes, int n_in,
                              void* d_out, int out_size, void* d_ws, size_t ws_size,
                              hipStream_t stream) {
  (void)n_in; (void)ws_size;
  const int B = 32;
  const int L = in_sizes[1] / B;                 // 1024
  const int D = in_sizes[0] / in_sizes[1];       // 512
  const int T = out_size / (B * D);              // 4096

  const float* x   = (const float*)d_in[0];
  const int*   dur = (const int*)d_in[1];        // harness: integer -> int32
  int*   cum = (int*)d_ws;                       // B*L*4 = 128 KB scratch
  float* out = (float*)d_out;

  lr_scan_kernel<<<B, L, 0, stream>>>(dur, cum, L);
  lr_gather_kernel<<<dim3(T, B), D / 4, 0, stream>>>(x, cum, out, L, D, T);
}